// VectorQuantize_30889404792944
// MI455X (gfx1250) — compile-verified
//
#include <hip/hip_runtime.h>
#include <hip/hip_bf16.h>
#include <float.h>

typedef __attribute__((ext_vector_type(2))) float v2f;
typedef __attribute__((ext_vector_type(8))) float v8f;

#define NN 32768
#define DD 128
#define KK 4096
#define DECAY 0.8f
#define EPSV 1e-5f
#define COMMIT_W 0.25f

// workspace layout (4-byte units)
#define WS_IDX   0           // int32 [NN]
#define WS_BINS  32768       // float [KK]
#define WS_ESUM  36864       // float [KK*DD]
#define WS_E2    561152      // float [KK]
#define WS_LOSS  565248      // float [1]
#define WS_NTOT  565249      // float [1]
#define WS_ZERO_CNT (565250 - WS_BINS)

// output layout (float units)
#define O_Q    0
#define O_IND  4194304
#define O_LOSS 4227072
#define O_CS   4227073
#define O_NORM 4231169

// ---------------- kernel 1: zero the accumulators ----------------
__global__ void vq_init(float* ws) {
    int g = blockIdx.x * blockDim.x + threadIdx.x;
    if (g < WS_ZERO_CNT) ws[WS_BINS + g] = 0.0f;
}

// ---------------- kernel 2: e2[k] = sum_d e[k,d]^2 ----------------
__global__ void vq_e2(const float* __restrict__ e, float* __restrict__ e2) {
    int lane = threadIdx.x & 31;
    int row  = blockIdx.x * 8 + (threadIdx.x >> 5);
    const float4* r = (const float4*)(e + (size_t)row * DD);
    float4 v = r[lane];
    float s = v.x * v.x + v.y * v.y + v.z * v.z + v.w * v.w;
    #pragma unroll
    for (int m = 16; m; m >>= 1) s += __shfl_xor(s, m, 32);
    if (lane == 0) e2[row] = s;
}

// ---------------- kernel 3: fused GEMM (f32 WMMA) + argmax ----------------
// block = 256 threads = 8 waves; each wave owns 32 rows of x (two 16-row A tiles
// feeding two independent accumulator chains per B fetch -> 2 WMMAs per LDS v2f).
// Block covers 256 rows; loops over K in tiles of 64 codebook rows staged in LDS.
__global__ __launch_bounds__(256) void vq_argmax(const float* __restrict__ x,
                                                 const float* __restrict__ e,
                                                 const float* __restrict__ e2,
                                                 int* __restrict__ idx_out,
                                                 float* __restrict__ indf_out) {
    __shared__ float lds_e[64 * 132];   // 64 rows, stride 132 floats (528B, 16B aligned, conflict-free)
    __shared__ float lds_e2[64];

    const int tid   = threadIdx.x;
    const int wave  = tid >> 5;
    const int lane  = tid & 31;
    const int l15   = lane & 15;
    const int lhalf = lane >> 4;            // 0 or 1
    const int row_base = blockIdx.x * 256 + wave * 32;
    const int m0 = row_base + l15;          // A-tile 0 row this lane supplies
    const int m1 = m0 + 16;                 // A-tile 1 row

    // A operands: two 16x128 x-tiles * 2.0, resident in 128 VGPRs per lane.
    // 16x4 f32 A layout: lane<16 -> K = 4*kk + {0,1}; lane>=16 -> K = 4*kk + {2,3}
    v2f a0[32], a1[32];
    {
        const v2f* xr0 = (const v2f*)(x + (size_t)m0 * DD);
        const v2f* xr1 = (const v2f*)(x + (size_t)m1 * DD);
        #pragma unroll
        for (int kk = 0; kk < 32; ++kk) {
            a0[kk] = xr0[kk * 2 + lhalf] * 2.0f;
            a1[kk] = xr1[kk * 2 + lhalf] * 2.0f;
        }
    }

    float best0[8], best1[8];
    int   bidx0[8], bidx1[8];
    #pragma unroll
    for (int v = 0; v < 8; ++v) {
        best0[v] = -FLT_MAX; bidx0[v] = 0;
        best1[v] = -FLT_MAX; bidx1[v] = 0;
    }

    for (int kt = 0; kt < KK / 64; ++kt) {
        __syncthreads();
        // cooperative stage: 64 rows x 128 floats (float4 granularity, padded rows)
        const float4* esrc = (const float4*)(e + (size_t)kt * 64 * DD);
        #pragma unroll
        for (int j = 0; j < 8; ++j) {
            int seg  = tid + j * 256;        // 2048 float4 segments total
            int row  = seg >> 5;             // 32 segments per row
            int col4 = seg & 31;
            *(float4*)(lds_e + row * 132 + col4 * 4) = esrc[row * 32 + col4];
        }
        if (tid < 64) lds_e2[tid] = e2[kt * 64 + tid];
        __syncthreads();

        #pragma unroll
        for (int sub = 0; sub < 4; ++sub) {
            const int n_l = sub * 16 + l15;
            // B operand: 4x16 tile of e^T; B[r,c] = e_tile[c][4*kk + r]
            const v2f* brow = (const v2f*)(lds_e + n_l * 132);
            v8f c0 = {};
            v8f c1 = {};
            #pragma unroll
            for (int kk = 0; kk < 32; ++kk) {
                v2f b = brow[kk * 2 + lhalf];
                c0 = __builtin_amdgcn_wmma_f32_16x16x4_f32(
                         false, a0[kk], false, b, (short)0, c0, false, false);
                c1 = __builtin_amdgcn_wmma_f32_16x16x4_f32(
                         false, a1[kk], false, b, (short)0, c1, false, false);
            }
            const float e2v = lds_e2[n_l];
            const int   ngl = kt * 64 + n_l;
            #pragma unroll
            for (int v = 0; v < 8; ++v) {
                float s0 = c0[v] - e2v;      // 2*x.e - |e|^2 (x^2 constant per row)
                float s1 = c1[v] - e2v;
                if (s0 > best0[v]) { best0[v] = s0; bidx0[v] = ngl; }  // ascending scan keeps first max
                if (s1 > best1[v]) { best1[v] = s1; bidx1[v] = ngl; }
            }
        }
    }

    // reduce across the 16 lanes holding the same output rows (column groups)
    #pragma unroll
    for (int v = 0; v < 8; ++v) {
        float b = best0[v]; int i = bidx0[v];
        #pragma unroll
        for (int m = 1; m < 16; m <<= 1) {
            float ob = __shfl_xor(b, m, 16);
            int   oi = __shfl_xor(i, m, 16);
            if (ob > b || (ob == b && oi < i)) { b = ob; i = oi; }
        }
        bidx0[v] = i;
        b = best1[v]; i = bidx1[v];
        #pragma unroll
        for (int m = 1; m < 16; m <<= 1) {
            float ob = __shfl_xor(b, m, 16);
            int   oi = __shfl_xor(i, m, 16);
            if (ob > b || (ob == b && oi < i)) { b = ob; i = oi; }
        }
        bidx1[v] = i;
    }
    if (l15 == 0) {
        #pragma unroll
        for (int v = 0; v < 8; ++v) {
            int row = row_base + v + 8 * lhalf;   // C/D layout: M = v + 8*(lane>=16)
            idx_out[row]       = bidx0[v];
            indf_out[row]      = (float)bidx0[v];
            idx_out[row + 16]  = bidx1[v];
            indf_out[row + 16] = (float)bidx1[v];
        }
    }
}

// ---------------- kernel 4: gather quantize, scatter EMA stats, loss partial ----------------
__global__ void vq_scatter(const float* __restrict__ x,
                           const float* __restrict__ e,
                           const int* __restrict__ idx,
                           float* __restrict__ q_out,
                           float* __restrict__ bins,
                           float* __restrict__ esum,
                           float* __restrict__ loss) {
    __shared__ float red[128];
    const int n = blockIdx.x;
    const int d = threadIdx.x;
    const int k = idx[n];
    float qv = e[(size_t)k * DD + d];
    float xv = x[(size_t)n * DD + d];
    q_out[(size_t)n * DD + d] = qv;
    float diff = qv - xv;
    red[d] = diff * diff;
    __syncthreads();
    #pragma unroll
    for (int s = 64; s > 0; s >>= 1) {
        if (d < s) red[d] += red[d + s];
        __syncthreads();
    }
    atomicAdd(&esum[(size_t)k * DD + d], xv);
    if (d == 0) {
        atomicAdd(loss, red[0]);
        atomicAdd(&bins[k], 1.0f);
    }
}

// ---------------- kernel 5a: cs_new + n_total ----------------
__global__ void vq_cs(const float* __restrict__ cluster_size,
                      const float* __restrict__ bins,
                      float* __restrict__ cs_out,
                      float* __restrict__ ntot) {
    __shared__ float red[256];
    int k = blockIdx.x * 256 + threadIdx.x;
    float cs = cluster_size[k] * DECAY + (1.0f - DECAY) * bins[k];
    cs_out[k] = cs;
    red[threadIdx.x] = cs;
    __syncthreads();
    #pragma unroll
    for (int s = 128; s > 0; s >>= 1) {
        if (threadIdx.x < s) red[threadIdx.x] += red[threadIdx.x + s];
        __syncthreads();
    }
    if (threadIdx.x == 0) atomicAdd(ntot, red[0]);
}

// ---------------- kernel 5b: embed_normalized + final loss ----------------
__global__ void vq_norm(const float* __restrict__ embed_avg,
                        const float* __restrict__ esum,
                        const float* __restrict__ cs_out,
                        const float* __restrict__ ntot,
                        const float* __restrict__ loss_sum,
                        float* __restrict__ norm_out,
                        float* __restrict__ loss_out) {
    int g = blockIdx.x * 256 + threadIdx.x;
    int k = g >> 7;
    float cs = cs_out[k];
    float nt = *ntot;
    float cluster = (cs + EPSV) / (nt + (float)KK * EPSV) * nt;
    float ean = embed_avg[g] * DECAY + (1.0f - DECAY) * esum[g];
    norm_out[g] = ean / cluster;
    if (g == 0) loss_out[0] = loss_sum[0] * COMMIT_W / (float)((size_t)NN * DD);
}

extern "C" void kernel_launch(void* const* d_in, const int* in_sizes, int n_in,
                              void* d_out, int out_size, void* d_ws, size_t ws_size,
                              hipStream_t stream) {
    const float* x            = (const float*)d_in[0];
    const float* embed        = (const float*)d_in[1];   // [1,K,D]
    const float* cluster_size = (const float*)d_in[2];   // [1,K]
    const float* embed_avg    = (const float*)d_in[3];   // [1,K,D]

    float* out  = (float*)d_out;
    float* ws   = (float*)d_ws;
    int*   wsid = (int*)d_ws;

    float* q_out    = out + O_Q;
    float* indf_out = out + O_IND;
    float* loss_out = out + O_LOSS;
    float* cs_out   = out + O_CS;
    float* norm_out = out + O_NORM;

    int*   idx_ws  = wsid + WS_IDX;
    float* bins    = ws + WS_BINS;
    float* esum    = ws + WS_ESUM;
    float* e2      = ws + WS_E2;
    float* loss_ws = ws + WS_LOSS;
    float* ntot    = ws + WS_NTOT;

    vq_init<<<(WS_ZERO_CNT + 255) / 256, 256, 0, stream>>>(ws);
    vq_e2<<<KK / 8, 256, 0, stream>>>(embed, e2);
    vq_argmax<<<NN / 256, 256, 0, stream>>>(x, embed, e2, idx_ws, indf_out);
    vq_scatter<<<NN, 128, 0, stream>>>(x, embed, idx_ws, q_out, bins, esum, loss_ws);
    vq_cs<<<KK / 256, 256, 0, stream>>>(cluster_size, bins, cs_out, ntot);
    vq_norm<<<(KK * DD) / 256, 256, 0, stream>>>(embed_avg, esum, cs_out, ntot,
                                                 loss_ws, norm_out, loss_out);
}